// ShiftNormBatch2d_13786845020238
// MI455X (gfx1250) — compile-verified
//
#include <hip/hip_runtime.h>
#include <math.h>

// ShiftNormBatch2d for x[N=64, C=128, H=56, W=56] fp32.
// Memory-bound: 3 reads of x (2nd/3rd hit the 192MB L2) + 1 write.
// Per-channel reductions use V_WMMA_F32_16X16X4_F32 as a ones-matmul reducer
// (f32 precision; the only f32 WMMA shape on CDNA5). Output pass uses
// non-temporal load/store so the out stream does not evict x from L2.

typedef float v2f __attribute__((ext_vector_type(2)));
typedef float v4f __attribute__((ext_vector_type(4)));
typedef float v8f __attribute__((ext_vector_type(8)));

#define NB   64
#define CB   128
#define HWE  3136          // H*W
#define HW4  784           // H*W / 4 (float4 count per (n,c) slab)
#define NHWf 200704.0f     // N*H*W
#define EPSV 1e-5f

// AP2(x) = sign(x) * 2^round(log2|x|).
// round(log2(1.m)) == 1 iff 1.m > sqrt(2)  => bump exponent when mantissa > 0x3504F3.
__device__ __forceinline__ float ap2f(float x) {
    unsigned u  = __float_as_uint(x);
    unsigned s  = u & 0x80000000u;
    unsigned ax = u & 0x7fffffffu;
    if (ax == 0u) return 0.0f;               // sign(0)*... = 0
    unsigned e = ax >> 23;
    unsigned m = ax & 0x7fffffu;
    e += (m > 0x3504f3u) ? 1u : 0u;
    return __uint_as_float(s | (e << 23));
}

// With B = all-ones, D[m][n] = sum_k A[m][k] (independent of n). Lane L holds
// rows 0-7 of column L, lane L+16 holds rows 8-15 of column L: per-lane sum of
// 8 + xor-16 shuffle-add = grand total of every element fed into A, all lanes.
__device__ __forceinline__ float wave_total(v8f c) {
    float s = c[0] + c[1] + c[2] + c[3] + c[4] + c[5] + c[6] + c[7];
    s += __shfl_xor(s, 16, 32);
    return s;
}

__device__ __forceinline__ void wmma_acc2(v4f v, v8f& acc0, v8f& acc1) {
    const v2f ones = {1.0f, 1.0f};
    v2f a0 = {v[0], v[1]};
    v2f a1 = {v[2], v[3]};
    // Two independent accumulators: halves the WMMA->WMMA D->C RAW chain.
    acc0 = __builtin_amdgcn_wmma_f32_16x16x4_f32(false, a0, false, ones, (short)0, acc0, false, false);
    acc1 = __builtin_amdgcn_wmma_f32_16x16x4_f32(false, a1, false, ones, (short)0, acc1, false, false);
}

// --- pass 1: per-(n,c) slab sums of x -------------------------------------
__global__ __launch_bounds__(256) void k_mean_partial(const float* __restrict__ x,
                                                      float* __restrict__ pm) {
    const int b = blockIdx.x;            // b = n*CB + c  (slab index)
    const int c = b % CB;
    const int n = b / CB;
    const v4f* __restrict__ x4 = (const v4f*)x + (size_t)b * HW4;
    const int tid = threadIdx.x;

    v8f acc0 = {}, acc1 = {};
    for (int base = 0; base < HW4; base += 256) {     // uniform bound: EXEC all-1s at WMMA
        const int i = base + tid;
        v4f v = {0.f, 0.f, 0.f, 0.f};
        if (i < HW4) v = x4[i];                        // zero-fill guard, not lane-mask
        wmma_acc2(v, acc0, acc1);
    }
    const float wt = wave_total(acc0 + acc1);

    __shared__ float wsum[8];
    const int lane = tid & 31, wid = tid >> 5;
    if (lane == 0) wsum[wid] = wt;
    __syncthreads();
    if (tid == 0) {
        float s = 0.f;
        #pragma unroll
        for (int i = 0; i < 8; ++i) s += wsum[i];
        pm[(size_t)c * NB + n] = s;
    }
}

// --- finalize mean: rm = 0.9*running_mean + 0.1*batch_mean ----------------
__global__ void k_fin_mean(const float* __restrict__ pm,
                           const float* __restrict__ running_mean,
                           float* __restrict__ rm) {
    const int c = threadIdx.x;           // 128 threads
    float s = 0.f;
    for (int n = 0; n < NB; ++n) s += pm[(size_t)c * NB + n];
    const float mean = s * (1.0f / NHWf);
    rm[c] = 0.9f * running_mean[c] + 0.1f * mean;
}

// --- pass 2: per-(n,c) slab sums of (x-rm)*AP2(x-rm) ----------------------
__global__ __launch_bounds__(256) void k_var_partial(const float* __restrict__ x,
                                                     const float* __restrict__ rm,
                                                     float* __restrict__ pv) {
    const int b = blockIdx.x;
    const int c = b % CB;
    const int n = b / CB;
    const float m = rm[c];
    const v4f* __restrict__ x4 = (const v4f*)x + (size_t)b * HW4;
    const int tid = threadIdx.x;

    v8f acc0 = {}, acc1 = {};
    for (int base = 0; base < HW4; base += 256) {
        const int i = base + tid;
        const bool ok = (i < HW4);
        v4f v = {0.f, 0.f, 0.f, 0.f};
        if (ok) v = x4[i];
        v4f q;
        #pragma unroll
        for (int k = 0; k < 4; ++k) {
            const float t = v[k] - m;
            q[k] = ok ? t * ap2f(t) : 0.f;   // OOB lanes must contribute 0 (t=-m otherwise)
        }
        wmma_acc2(q, acc0, acc1);
    }
    const float wt = wave_total(acc0 + acc1);

    __shared__ float wsum[8];
    const int lane = tid & 31, wid = tid >> 5;
    if (lane == 0) wsum[wid] = wt;
    __syncthreads();
    if (tid == 0) {
        float s = 0.f;
        #pragma unroll
        for (int i = 0; i < 8; ++i) s += wsum[i];
        pv[(size_t)c * NB + n] = s;
    }
}

// --- finalize var: scale = AP2(weight) * AP2(1/sqrt(rv+eps)) --------------
__global__ void k_fin_var(const float* __restrict__ pv,
                          const float* __restrict__ running_var,
                          const float* __restrict__ weight,
                          float* __restrict__ scale) {
    const int c = threadIdx.x;           // 128 threads
    float s = 0.f;
    for (int n = 0; n < NB; ++n) s += pv[(size_t)c * NB + n];
    const float var = s * (1.0f / NHWf);
    const float rv = 0.9f * running_var[c] + 0.1f * var;
    const float inv_std = ap2f(1.0f / sqrtf(rv + EPSV));
    scale[c] = ap2f(weight[c]) * inv_std;
}

// --- pass 3: out = scale[c]*(x - rm[c]) + bias[c] -------------------------
// Non-temporal on both streams: x is on its last read (let L2 retire it),
// and out must not evict x's lines during the pass (x+out > 192MB L2).
__global__ __launch_bounds__(256) void k_out(const float* __restrict__ x,
                                             const float* __restrict__ rm,
                                             const float* __restrict__ scale,
                                             const float* __restrict__ bias,
                                             float* __restrict__ out) {
    const size_t i = (size_t)blockIdx.x * 256 + threadIdx.x;   // float4 index
    const int c = (int)((i / HW4) % CB);                       // all 4 elems same channel
    const float m = rm[c], sc = scale[c], bi = bias[c];
    const v4f v = __builtin_nontemporal_load((const v4f*)x + i);
    v4f r;
    #pragma unroll
    for (int k = 0; k < 4; ++k) r[k] = fmaf(sc, v[k] - m, bi);
    __builtin_nontemporal_store(r, (v4f*)out + i);
}

extern "C" void kernel_launch(void* const* d_in, const int* in_sizes, int n_in,
                              void* d_out, int out_size, void* d_ws, size_t ws_size,
                              hipStream_t stream) {
    (void)in_sizes; (void)n_in; (void)out_size; (void)ws_size;
    const float* x            = (const float*)d_in[0];
    const float* weight       = (const float*)d_in[1];
    const float* bias         = (const float*)d_in[2];
    const float* running_mean = (const float*)d_in[3];
    const float* running_var  = (const float*)d_in[4];
    float* out = (float*)d_out;

    float* pm    = (float*)d_ws;          // [CB*NB] = 8192 floats
    float* pv    = pm + CB * NB;          // [CB*NB]
    float* rm    = pv + CB * NB;          // [CB]
    float* scale = rm + CB;               // [CB]

    k_mean_partial<<<NB * CB, 256, 0, stream>>>(x, pm);
    k_fin_mean   <<<1, CB, 0, stream>>>(pm, running_mean, rm);
    k_var_partial<<<NB * CB, 256, 0, stream>>>(x, rm, pv);
    k_fin_var    <<<1, CB, 0, stream>>>(pv, running_var, weight, scale);
    // 25,690,112 elems / 4 = 6,422,528 float4 = 25088 blocks * 256 threads exactly
    k_out        <<<25088, 256, 0, stream>>>(x, rm, scale, bias, out);
}